// ANI2xt_33681133535926
// MI455X (gfx1250) — compile-verified
//
#include <hip/hip_runtime.h>
#include <hip/hip_bf16.h>

// ---------------------------------------------------------------------------
// ANI-2x energy + force on gfx1250 (MI455X).
// fp32 end-to-end, MLP GEMMs via V_WMMA_F32_16X16X4_F32 (wave32 tensor op).
// ---------------------------------------------------------------------------

#define H2EV  27.211386024367243f
#define RCRc  5.2f
#define RCAc  3.5f
#define PI_F  3.14159265358979323846f
#define NSPEC 7

typedef __attribute__((ext_vector_type(2))) float v2f;
typedef __attribute__((ext_vector_type(8))) float v8f;

__constant__ float c_shfr[16] = {0.9f,1.16875f,1.4375f,1.70625f,1.975f,2.24375f,2.5125f,2.78125f,
                                 3.05f,3.31875f,3.5875f,3.85625f,4.125f,4.39375f,4.6625f,4.93125f};
__constant__ float c_shfz[8]  = {0.19634954f,0.58904862f,0.9817477f,1.3744468f,
                                 1.7671459f,2.1598449f,2.552544f,2.9452431f};
__constant__ float c_shfa[4]  = {0.9f,1.55f,2.2f,2.85f};

__device__ __forceinline__ int triu_idx(int a, int b) {
  if (a > b) { int t = a; a = b; b = t; }
  return a * NSPEC - (a * (a - 1)) / 2 + (b - a);
}

__device__ __forceinline__ float celu_f(float x) {
  return x > 0.f ? x : 0.1f * (__expf(10.f * x) - 1.f);
}
__device__ __forceinline__ float celu_p(float x) {
  return x > 0.f ? 1.f : __expf(10.f * x);
}

__device__ __forceinline__ v8f wmma_f32(v2f a, v2f b, v8f c) {
  // D(16x16,f32) = A(16x4,f32) * B(4x16,f32) + C
  return __builtin_amdgcn_wmma_f32_16x16x4_f32(
      /*neg_a=*/false, a, /*neg_b=*/false, b,
      /*c_mod=*/(short)0, c, /*reuse_a=*/false, /*reuse_b=*/false);
}

// ---------------------------------------------------------------------------
// Kernel 1: AEV forward.  One block (128 threads) per atom.
//   aev layout: [4096][1024] (only [0,1008) valid)
// ---------------------------------------------------------------------------
__global__ void __launch_bounds__(128)
aev_kernel(const int* __restrict__ species, const float* __restrict__ coords,
           float* __restrict__ aev)
{
  const int idx = blockIdx.x;            // 0..4095
  const int b = idx >> 5, i = idx & 31;
  const int tid = threadIdx.x;

  __shared__ float svx[32], svy[32], svz[32], sd[32], sfcr[32], sfca[32];
  __shared__ int   ssp[32];
  __shared__ float acc[1008];

  for (int t = tid; t < 1008; t += 128) acc[t] = 0.f;
  if (tid < 32) {
    int j = tid;
    float ix = coords[(b*32+i)*3+0], iy = coords[(b*32+i)*3+1], iz = coords[(b*32+i)*3+2];
    float dx = coords[(b*32+j)*3+0] - ix;
    float dy = coords[(b*32+j)*3+1] - iy;
    float dz = coords[(b*32+j)*3+2] - iz;
    float d  = sqrtf(dx*dx + dy*dy + dz*dz);
    svx[j] = dx; svy[j] = dy; svz[j] = dz; sd[j] = d;
    ssp[j] = species[b*32+j];
    bool ok = (j != i);
    sfcr[j] = (ok && d < RCRc) ? (0.5f*__cosf(PI_F*d/RCRc) + 0.5f) : 0.f;
    sfca[j] = (ok && d < RCAc) ? (0.5f*__cosf(PI_F*d/RCAc) + 0.5f) : 0.f;
  }
  __syncthreads();

  // radial: 32 neighbors x 16 shifts
  for (int p = tid; p < 512; p += 128) {
    int j = p >> 4, t = p & 15;
    float fc = sfcr[j];
    if (j != i && fc > 0.f) {
      float dd = sd[j] - c_shfr[t];
      atomicAdd(&acc[ssp[j]*16 + t], 0.25f * __expf(-16.f*dd*dd) * fc);
    }
  }

  // angular: ordered pairs (j,k)
  for (int p = tid; p < 1024; p += 128) {
    int j = p >> 5, k = p & 31;
    if (j == k || j == i || k == i) continue;
    float fj = sfca[j], fk = sfca[k];
    float w = fj * fk;
    if (w <= 0.f) continue;
    float du = sd[j], dv = sd[k];
    float dot = svx[j]*svx[k] + svy[j]*svy[k] + svz[j]*svz[k];
    float craw = 0.95f * dot / (du * dv);
    float cc = fminf(0.95f, fmaxf(-0.95f, craw));
    float th = acosf(cc);
    float ravg = 0.5f * (du + dv);
    int pr = triu_idx(ssp[j], ssp[k]);
    float f2v[4];
    #pragma unroll
    for (int a = 0; a < 4; ++a) { float r = ravg - c_shfa[a]; f2v[a] = __expf(-8.f*r*r); }
    #pragma unroll
    for (int z = 0; z < 8; ++z) {
      float t1 = 0.5f * (1.f + __cosf(th - c_shfz[z]));
      float t2 = t1*t1, t4 = t2*t2, t8 = t4*t4, t16 = t8*t8;
      float base = t16 * t16 * w;   // t1^32 * fj*fk
      #pragma unroll
      for (int a = 0; a < 4; ++a)
        atomicAdd(&acc[112 + pr*32 + z*4 + a], base * f2v[a]);
    }
  }
  __syncthreads();
  for (int t = tid; t < 1008; t += 128) aev[(size_t)idx*1024 + t] = acc[t];
}

// ---------------------------------------------------------------------------
// Kernel 2: fused MLP forward + backward-to-AEV for one species.
//   Block = 128 threads (4 wave32s), one 16-atom tile.  All GEMM tiles via
//   v_wmma_f32_16x16x4_f32; pre-activations kept in LDS; dz computed in place.
// ---------------------------------------------------------------------------
template<int H1, int H2, int H3>
__global__ void __launch_bounds__(128)
mlp_kernel(const float* __restrict__ aev, const int* __restrict__ species, int sidx,
           const float* __restrict__ W1, const float* __restrict__ B1,
           const float* __restrict__ W2, const float* __restrict__ B2,
           const float* __restrict__ W3, const float* __restrict__ B3,
           const float* __restrict__ W4, const float* __restrict__ B4,
           float* __restrict__ e_at, float* __restrict__ dEdaev)
{
  __shared__ float z1[16 * H1];
  __shared__ float z2[16 * H2];
  __shared__ float z3[16 * H3];
  __shared__ int   ssp[16];

  const int tid  = threadIdx.x;
  const int lane = tid & 31;
  const int wave = tid >> 5;
  const int row  = lane & 15;     // A-frag row / D-frag column index
  const int hi   = lane >> 4;     // 0 = low half, 1 = high half of wave
  const int base = blockIdx.x * 16;

  if (tid < 16) ssp[tid] = species[base + tid];
  __syncthreads();

  // ---- layer 1: z1 = aev @ W1 + b1   (K = 1008, N = H1) ----
  {
    const float* arow = aev + (size_t)(base + row) * 1024;
    for (int nt = wave * 16; nt < H1; nt += 64) {
      v8f c = {0.f,0.f,0.f,0.f,0.f,0.f,0.f,0.f};
      const int col = nt + row;
      for (int kb = 0; kb < 1008; kb += 4) {
        int k0 = kb + (hi << 1);
        v2f a;  a.x  = arow[k0];                       a.y  = arow[k0 + 1];
        v2f bb; bb.x = W1[(size_t)k0 * H1 + col];      bb.y = W1[(size_t)(k0 + 1) * H1 + col];
        c = wmma_f32(a, bb, c);
      }
      #pragma unroll
      for (int v = 0; v < 8; ++v)
        z1[(v + (hi << 3)) * H1 + col] = c[v] + B1[col];
    }
  }
  __syncthreads();

  // ---- layer 2: z2 = celu(z1) @ W2 + b2 ----
  for (int nt = wave * 16; nt < H2; nt += 64) {
    v8f c = {0.f,0.f,0.f,0.f,0.f,0.f,0.f,0.f};
    const int col = nt + row;
    for (int kb = 0; kb < H1; kb += 4) {
      int k0 = kb + (hi << 1);
      v2f a;  a.x  = celu_f(z1[row * H1 + k0]);  a.y  = celu_f(z1[row * H1 + k0 + 1]);
      v2f bb; bb.x = W2[k0 * H2 + col];          bb.y = W2[(k0 + 1) * H2 + col];
      c = wmma_f32(a, bb, c);
    }
    #pragma unroll
    for (int v = 0; v < 8; ++v)
      z2[(v + (hi << 3)) * H2 + col] = c[v] + B2[col];
  }
  __syncthreads();

  // ---- layer 3: z3 = celu(z2) @ W3 + b3 ----
  for (int nt = wave * 16; nt < H3; nt += 64) {
    v8f c = {0.f,0.f,0.f,0.f,0.f,0.f,0.f,0.f};
    const int col = nt + row;
    for (int kb = 0; kb < H2; kb += 4) {
      int k0 = kb + (hi << 1);
      v2f a;  a.x  = celu_f(z2[row * H2 + k0]);  a.y  = celu_f(z2[row * H2 + k0 + 1]);
      v2f bb; bb.x = W3[k0 * H3 + col];          bb.y = W3[(k0 + 1) * H3 + col];
      c = wmma_f32(a, bb, c);
    }
    #pragma unroll
    for (int v = 0; v < 8; ++v)
      z3[(v + (hi << 3)) * H3 + col] = c[v] + B3[col];
  }
  __syncthreads();

  // ---- output layer: e = celu(z3) @ W4 + b4 (masked store) ----
  if (tid < 16) {
    float e = B4[0];
    for (int n = 0; n < H3; ++n) e += celu_f(z3[tid * H3 + n]) * W4[n];
    if (ssp[tid] == sidx) e_at[base + tid] = e;
  }
  __syncthreads();

  // ---- backward: dz3 = W4 * celu'(z3)   (in place) ----
  for (int p = tid; p < 16 * H3; p += 128)
    z3[p] = W4[p % H3] * celu_p(z3[p]);
  __syncthreads();

  // ---- dh2 = dz3 @ W3^T ; dz2 = dh2 * celu'(z2)  (in place into z2) ----
  for (int nt = wave * 16; nt < H2; nt += 64) {
    v8f c = {0.f,0.f,0.f,0.f,0.f,0.f,0.f,0.f};
    const int col = nt + row;
    for (int kb = 0; kb < H3; kb += 4) {
      int k0 = kb + (hi << 1);
      v2f a;  a.x  = z3[row * H3 + k0];       a.y  = z3[row * H3 + k0 + 1];
      v2f bb; bb.x = W3[col * H3 + k0];       bb.y = W3[col * H3 + k0 + 1];   // W3^T
      c = wmma_f32(a, bb, c);
    }
    #pragma unroll
    for (int v = 0; v < 8; ++v) {
      int r = v + (hi << 3);
      z2[r * H2 + col] = c[v] * celu_p(z2[r * H2 + col]);
    }
  }
  __syncthreads();

  // ---- dh1 = dz2 @ W2^T ; dz1 = dh1 * celu'(z1)  (in place into z1) ----
  for (int nt = wave * 16; nt < H1; nt += 64) {
    v8f c = {0.f,0.f,0.f,0.f,0.f,0.f,0.f,0.f};
    const int col = nt + row;
    for (int kb = 0; kb < H2; kb += 4) {
      int k0 = kb + (hi << 1);
      v2f a;  a.x  = z2[row * H2 + k0];       a.y  = z2[row * H2 + k0 + 1];
      v2f bb; bb.x = W2[col * H2 + k0];       bb.y = W2[col * H2 + k0 + 1];   // W2^T
      c = wmma_f32(a, bb, c);
    }
    #pragma unroll
    for (int v = 0; v < 8; ++v) {
      int r = v + (hi << 3);
      z1[r * H1 + col] = c[v] * celu_p(z1[r * H1 + col]);
    }
  }
  __syncthreads();

  // ---- dAEV = dz1 @ W1^T  (K = H1, N = 1008), masked store per row ----
  for (int nt = wave * 16; nt < 1008; nt += 64) {
    v8f c = {0.f,0.f,0.f,0.f,0.f,0.f,0.f,0.f};
    const int col = nt + row;
    for (int kb = 0; kb < H1; kb += 4) {
      int k0 = kb + (hi << 1);
      v2f a;  a.x  = z1[row * H1 + k0];              a.y  = z1[row * H1 + k0 + 1];
      v2f bb; bb.x = W1[(size_t)col * H1 + k0];      bb.y = W1[(size_t)col * H1 + k0 + 1]; // W1^T
      c = wmma_f32(a, bb, c);
    }
    #pragma unroll
    for (int v = 0; v < 8; ++v) {
      int r = v + (hi << 3);
      if (ssp[r] == sidx)
        dEdaev[(size_t)(base + r) * 1024 + col] = c[v];
    }
  }
}

// ---------------------------------------------------------------------------
// Kernel 3: per-molecule energy reduction (wave32 shuffle).
// ---------------------------------------------------------------------------
__global__ void __launch_bounds__(32)
energy_kernel(const float* __restrict__ e_at, const int* __restrict__ species,
              const float* __restrict__ selfE, float* __restrict__ out)
{
  int b = blockIdx.x, t = threadIdx.x;
  float e = e_at[b*32 + t] + selfE[species[b*32 + t]];
  #pragma unroll
  for (int off = 16; off > 0; off >>= 1) e += __shfl_down(e, off, 32);
  if (t == 0) out[b] = e * H2EV;
}

// ---------------------------------------------------------------------------
// Kernel 4: AEV backward -> forces.  One block (256 threads) per molecule;
// all gradient accumulation in LDS (no global atomics).
// ---------------------------------------------------------------------------
__global__ void __launch_bounds__(256)
force_kernel(const int* __restrict__ species, const float* __restrict__ coords,
             const float* __restrict__ dEdaev, float* __restrict__ force)
{
  const int b = blockIdx.x;
  const int tid = threadIdx.x;
  __shared__ float px[32], py[32], pz[32];
  __shared__ float gx[32], gy[32], gz[32];
  __shared__ int   ssp[32];

  if (tid < 32) {
    px[tid] = coords[(b*32+tid)*3+0];
    py[tid] = coords[(b*32+tid)*3+1];
    pz[tid] = coords[(b*32+tid)*3+2];
    ssp[tid] = species[b*32+tid];
    gx[tid] = 0.f; gy[tid] = 0.f; gz[tid] = 0.f;
  }
  __syncthreads();

  // ---- radial gradient: pairs (i,j) ----
  for (int p = tid; p < 1024; p += 256) {
    int i = p >> 5, j = p & 31;
    if (i == j) continue;
    float ux = px[j]-px[i], uy = py[j]-py[i], uz = pz[j]-pz[i];
    float d = sqrtf(ux*ux + uy*uy + uz*uz);
    if (d >= RCRc) continue;
    const float* dE = dEdaev + (size_t)(b*32+i)*1024 + ssp[j]*16;
    float fc  = 0.5f*__cosf(PI_F*d/RCRc) + 0.5f;
    float dfc = -0.5f*(PI_F/RCRc)*__sinf(PI_F*d/RCRc);
    float g = 0.f;
    #pragma unroll
    for (int t = 0; t < 16; ++t) {
      float dd = d - c_shfr[t];
      float ex = __expf(-16.f*dd*dd);
      g += dE[t] * 0.25f * (ex*dfc - 32.f*dd*ex*fc);
    }
    float inv = 1.f / d;
    float hx = g*ux*inv, hy = g*uy*inv, hz = g*uz*inv;
    atomicAdd(&gx[j],  hx); atomicAdd(&gy[j],  hy); atomicAdd(&gz[j],  hz);
    atomicAdd(&gx[i], -hx); atomicAdd(&gy[i], -hy); atomicAdd(&gz[i], -hz);
  }

  // ---- angular gradient: ordered triples (i; j,k) ----
  for (int p = tid; p < 32768; p += 256) {
    int i = p >> 10, j = (p >> 5) & 31, k = p & 31;
    if (j == k || j == i || k == i) continue;
    float ux = px[j]-px[i], uy = py[j]-py[i], uz = pz[j]-pz[i];
    float vx = px[k]-px[i], vy = py[k]-py[i], vz = pz[k]-pz[i];
    float du = sqrtf(ux*ux + uy*uy + uz*uz);
    float dv = sqrtf(vx*vx + vy*vy + vz*vz);
    if (du >= RCAc || dv >= RCAc) continue;
    float fj  = 0.5f*__cosf(PI_F*du/RCAc) + 0.5f;
    float fk  = 0.5f*__cosf(PI_F*dv/RCAc) + 0.5f;
    float dfj = -0.5f*(PI_F/RCAc)*__sinf(PI_F*du/RCAc);
    float dfk = -0.5f*(PI_F/RCAc)*__sinf(PI_F*dv/RCAc);
    float dot = ux*vx + uy*vy + uz*vz;
    float invu = 1.f/du, invv = 1.f/dv;
    float craw = 0.95f * dot * invu * invv;
    float cc = fminf(0.95f, fmaxf(-0.95f, craw));
    float th = acosf(cc);
    float dthdc = (craw > -0.95f && craw < 0.95f)
                  ? (-1.f / sqrtf(fmaxf(1.f - cc*cc, 1e-12f))) : 0.f;
    float ravg = 0.5f * (du + dv);
    const float* dE = dEdaev + (size_t)(b*32+i)*1024 + 112
                    + triu_idx(ssp[j], ssp[k]) * 32;
    float f2v[4], df2v[4];
    #pragma unroll
    for (int a = 0; a < 4; ++a) {
      float r = ravg - c_shfa[a];
      float e = __expf(-8.f*r*r);
      f2v[a] = e; df2v[a] = -16.f*r*e;
    }
    float S1 = 0.f, S2 = 0.f, S3 = 0.f;
    #pragma unroll
    for (int z = 0; z < 8; ++z) {
      float ph = th - c_shfz[z];
      float t1 = 0.5f*(1.f + __cosf(ph));
      float t2 = t1*t1, t4 = t2*t2, t8 = t4*t4, t16 = t8*t8;
      float t31 = t16*t8*t4*t2*t1;
      float f1  = t31*t1;
      float df1 = -16.f*t31*__sinf(ph);
      #pragma unroll
      for (int a = 0; a < 4; ++a) {
        float w = dE[z*4 + a];
        S1 += w * f2v[a]  * df1;
        S2 += w * df2v[a] * f1;
        S3 += w * f2v[a]  * f1;
      }
    }
    float W  = fj * fk;
    float cu = S1 * W * dthdc * 0.95f * invu * invv;   // coefficient for grad of cos
    float au = 0.5f*S2*W + S3*dfj*fk;                  // along u-hat
    float av = 0.5f*S2*W + S3*fj*dfk;                  // along v-hat
    float guX = cu*(vx - dot*invu*invu*ux) + au*ux*invu;
    float guY = cu*(vy - dot*invu*invu*uy) + au*uy*invu;
    float guZ = cu*(vz - dot*invu*invu*uz) + au*uz*invu;
    float gvX = cu*(ux - dot*invv*invv*vx) + av*vx*invv;
    float gvY = cu*(uy - dot*invv*invv*vy) + av*vy*invv;
    float gvZ = cu*(uz - dot*invv*invv*vz) + av*vz*invv;
    atomicAdd(&gx[j], guX); atomicAdd(&gy[j], guY); atomicAdd(&gz[j], guZ);
    atomicAdd(&gx[k], gvX); atomicAdd(&gy[k], gvY); atomicAdd(&gz[k], gvZ);
    atomicAdd(&gx[i], -(guX+gvX)); atomicAdd(&gy[i], -(guY+gvY)); atomicAdd(&gz[i], -(guZ+gvZ));
  }
  __syncthreads();

  if (tid < 32) {
    force[(b*32+tid)*3+0] = -H2EV * gx[tid];
    force[(b*32+tid)*3+1] = -H2EV * gy[tid];
    force[(b*32+tid)*3+2] = -H2EV * gz[tid];
  }
}

// ---------------------------------------------------------------------------
// Launch.
//   d_in: [0]=species(int,4096) [1]=coords(f32,12288)
//         [2..57]=params W/b flattened (7 species x 4 layers x {W,b})
//         [58]=self_energies(f32,7)
//   d_out: energy[128] ++ force[128*32*3] (f32)
// ---------------------------------------------------------------------------
extern "C" void kernel_launch(void* const* d_in, const int* in_sizes, int n_in,
                              void* d_out, int out_size, void* d_ws, size_t ws_size,
                              hipStream_t stream) {
  (void)in_sizes; (void)n_in; (void)out_size; (void)ws_size;
  const int*   species = (const int*)  d_in[0];
  const float* coords  = (const float*)d_in[1];
  const float* selfE   = (const float*)d_in[58];
  float* out   = (float*)d_out;
  float* aev   = (float*)d_ws;                 // 4096 * 1024 f32 (16 MB)
  float* dEda  = aev  + (size_t)4096 * 1024;   // 4096 * 1024 f32 (16 MB)
  float* e_at  = dEda + (size_t)4096 * 1024;   // 4096 f32

  aev_kernel<<<4096, 128, 0, stream>>>(species, coords, aev);

  auto Wp = [&](int s, int l) { return (const float*)d_in[2 + s*8 + l*2]; };
  auto Bp = [&](int s, int l) { return (const float*)d_in[2 + s*8 + l*2 + 1]; };

  #define LAUNCH_MLP(S, A, Bq, C)                                              \
    mlp_kernel<A, Bq, C><<<256, 128, 0, stream>>>(aev, species, S,             \
        Wp(S,0), Bp(S,0), Wp(S,1), Bp(S,1), Wp(S,2), Bp(S,2), Wp(S,3), Bp(S,3),\
        e_at, dEda)

  LAUNCH_MLP(0, 256, 192, 160);   // H
  LAUNCH_MLP(1, 224, 192, 160);   // C
  LAUNCH_MLP(2, 192, 160, 128);   // N
  LAUNCH_MLP(3, 192, 160, 128);   // O
  LAUNCH_MLP(4, 160, 128,  96);   // F
  LAUNCH_MLP(5, 160, 128,  96);   // S
  LAUNCH_MLP(6, 160, 128,  96);   // Cl
  #undef LAUNCH_MLP

  energy_kernel<<<128, 32, 0, stream>>>(e_at, species, selfE, out);
  force_kernel<<<128, 256, 0, stream>>>(species, coords, dEda, out + 128);
}